// BasicConvolutionBlock_4037269258942
// MI455X (gfx1250) — compile-verified
//
#include <hip/hip_runtime.h>
#include <hip/hip_bf16.h>

#define INC   32
#define OUTC  64
#define K3    27
#define ROWS_PER_BLOCK 128   // 8 waves * 16 rows

typedef float v2f __attribute__((ext_vector_type(2)));
typedef float v8f __attribute__((ext_vector_type(8)));

// Sparse submanifold conv: out[n,:] = ReLU( sum_k feats[kmap[k,n],:] @ w[k,:,:] )
// fp32 WMMA (16x16x4) keeps reference precision; gathers served from L2-resident feats.
// B staged into LDS pair-interleaved so each fragment is a single ds_load_b64.
__global__ __launch_bounds__(256) void spconv3d_wmma_f32(
    const float* __restrict__ feats,   // [N, 32]
    const float* __restrict__ weight,  // [27, 32, 64]
    const int*   __restrict__ kmap,    // [27, N]  (N == sentinel/missing)
    float*       __restrict__ out,     // [N, 64]
    int N)
{
    // K-pair interleaved stage of w[k]: wlds[p*64 + c] = (w[2p][c], w[2p+1][c])
    __shared__ v2f wlds[(INC / 2) * OUTC];        // 1024 float2 = 8 KB

    const int tid  = threadIdx.x;
    const int lane = tid & 31;
    const int wave = tid >> 5;
    const int half = lane >> 4;                   // 0: lanes 0-15, 1: lanes 16-31
    const int col  = lane & 15;                   // N-index within 16-wide tile
    const int row0 = blockIdx.x * ROWS_PER_BLOCK + wave * 16;

    // 16x64 output tile per wave: 4 accumulators (f32 16x16 C/D layout)
    v8f acc[4] = {};

    // Row this lane gathers for the A-fragment (A row M = lane&15)
    const int  arow_id = row0 + col;
    const bool rvalid  = arow_id < N;
    const int  rclamp  = rvalid ? arow_id : 0;

    for (int k = 0; k < K3; ++k) {
        // ---- stage w[k] (32x64 fp32) into LDS, K-pair interleaved ----
        __syncthreads();
        {
            const float* wk = weight + k * (INC * OUTC);
            #pragma unroll
            for (int e = tid; e < (INC / 2) * OUTC; e += 256) {
                const int p  = e >> 6;            // K-pair index 0..15
                const int cc = e & 63;            // column 0..63
                v2f w2;
                w2.x = wk[(2 * p + 0) * OUTC + cc];
                w2.y = wk[(2 * p + 1) * OUTC + cc];
                wlds[e] = w2;
            }
        }
        __syncthreads();

        // ---- gather index for this lane's A row (branchless sentinel) ----
        int idx = kmap[k * N + rclamp];
        idx = rvalid ? idx : N;
        const bool  miss = ((unsigned)idx >= (unsigned)N);
        const float asc  = miss ? 0.0f : 1.0f;
        const float* arow = feats + (size_t)(miss ? 0 : idx) * INC;

        // ---- K=32 as 8 chunks of 4; 4 col-tiles of 16 ----
        #pragma unroll
        for (int c = 0; c < 8; ++c) {
            const int kb = c * 4 + half * 2;      // this lane's K base within chunk
            v2f a = *(const v2f*)(arow + kb);     // A[m][kb], A[m][kb+1]
            a.x *= asc;                           // zero missing neighbor (no EXEC change)
            a.y *= asc;

            const int p = c * 2 + half;           // K-pair index for this lane
            #pragma unroll
            for (int t = 0; t < 4; ++t) {
                // (B[kb][col], B[kb+1][col]) of col-tile t: one ds_load_b64
                v2f b = wlds[p * OUTC + t * 16 + col];
                // D = A(16x4,f32) x B(4x16,f32) + C(16x16,f32)
                acc[t] = __builtin_amdgcn_wmma_f32_16x16x4_f32(
                    /*neg_a=*/false, a, /*neg_b=*/false, b,
                    /*c_mod=*/(short)0, acc[t],
                    /*reuse_a=*/false, /*reuse_b=*/false);
            }
        }
    }

    // ---- ReLU + store (D layout: lane L, vgpr r -> M = r + 8*half, N = col) ----
    #pragma unroll
    for (int t = 0; t < 4; ++t) {
        #pragma unroll
        for (int r = 0; r < 8; ++r) {
            const int row = row0 + half * 8 + r;
            if (row < N)
                out[(size_t)row * OUTC + t * 16 + col] = fmaxf(acc[t][r], 0.0f);
        }
    }
}

extern "C" void kernel_launch(void* const* d_in, const int* in_sizes, int n_in,
                              void* d_out, int out_size, void* d_ws, size_t ws_size,
                              hipStream_t stream)
{
    const float* feats  = (const float*)d_in[0];   // [N,32] fp32
    const float* weight = (const float*)d_in[1];   // [27,32,64] fp32
    const int*   kmap   = (const int*)d_in[2];     // [27,N] int32
    float*       out    = (float*)d_out;           // [N,64] fp32

    const int N = in_sizes[0] / INC;
    const int blocks = (N + ROWS_PER_BLOCK - 1) / ROWS_PER_BLOCK;

    hipLaunchKernelGGL(spconv3d_wmma_f32, dim3(blocks), dim3(256), 0, stream,
                       feats, weight, kmap, out, N);
}